// EquivariantFlowMatchingModel_30270929502224
// MI455X (gfx1250) — compile-verified
//
#include <hip/hip_runtime.h>
#include <hip/hip_bf16.h>
#include <math.h>

typedef __attribute__((ext_vector_type(2))) float v2f;
typedef __attribute__((ext_vector_type(8))) float v8f;

#define HID 128
#define TYPE_HID 64
#define EMBED 256
#define NUM_TYPES 5

__device__ __forceinline__ float sigm(float x) { return 1.0f / (1.0f + __expf(-x)); }
__device__ __forceinline__ float silu(float x) { return x * sigm(x); }

// Async global->LDS copy of 16 bytes per lane (CDNA5, tracked by ASYNCcnt).
// lds_off is the 32-bit LDS byte address (low 32 bits of a shared pointer),
// gptr the per-lane global source address.
__device__ __forceinline__ void async_copy_b128(unsigned lds_off, const void* gptr) {
  asm volatile("global_load_async_to_lds_b128 %0, %1, off"
               :: "v"(lds_off), "v"(gptr)
               : "memory");
}
__device__ __forceinline__ void wait_async_le4() {
  asm volatile("s_wait_asynccnt 0x4" ::: "memory");
}
__device__ __forceinline__ void wait_async_le0() {
  asm volatile("s_wait_asynccnt 0x0" ::: "memory");
}

// ---------------------------------------------------------------------------
// K0: ctxproj[d] = sum_k ctx[k] * Wt1[(128+k)*64 + d]   (shared by all atoms)
// ---------------------------------------------------------------------------
__global__ void k_ctx(const float* __restrict__ ctx, const float* __restrict__ Wt1,
                      float* __restrict__ ctxproj) {
  int d = threadIdx.x; // 64 threads
  float s = 0.0f;
  for (int k = 0; k < EMBED; ++k) s += ctx[k] * Wt1[(HID + k) * TYPE_HID + d];
  ctxproj[d] = s;
}

// ---------------------------------------------------------------------------
// K1: h = af@W_embed + b_embed ; m = h@W_msg + b_msg ; good flag
// one block (128 threads) per atom
// ---------------------------------------------------------------------------
__global__ void k_embed(const float* __restrict__ af, const float* __restrict__ W_embed,
                        const float* __restrict__ b_embed, const float* __restrict__ W_msg,
                        const float* __restrict__ b_msg, float* __restrict__ h,
                        float* __restrict__ m, float* __restrict__ good) {
  __shared__ float sh[HID];
  int i = blockIdx.x, d = threadIdx.x;
  float hv = b_embed[d];
#pragma unroll
  for (int t = 0; t < NUM_TYPES; ++t) hv += af[i * NUM_TYPES + t] * W_embed[t * HID + d];
  sh[d] = hv;
  __syncthreads();
  float mv = b_msg[d];
  for (int k = 0; k < HID; ++k) mv += sh[k] * W_msg[k * HID + d];
  h[(size_t)i * HID + d] = hv;
  m[(size_t)i * HID + d] = mv;
  if (d == 0) good[i] = (af[i * NUM_TYPES] != 1.0f) ? 1.0f : 0.0f;
}

// ---------------------------------------------------------------------------
// K2: agg = a @ m, with a computed on the fly (edge MLP + sigmoid, zero diag).
// 256 threads = 8 waves per block, 16 i-rows per block.
// m chunks (32 x 128) are double-buffered in LDS via async global->LDS
// (global_load_async_to_lds_b128, ASYNCcnt), overlapping the edge-MLP and
// WMMA phases of the current chunk with the fetch of the next one.
// fp32 WMMA 16x16x4 does the 16x16 (rows x H-cols) accumulation per wave.
// ---------------------------------------------------------------------------
__global__ void k_attn(const float* __restrict__ pos, const float* __restrict__ We1,
                       const float* __restrict__ be1, const float* __restrict__ We2,
                       const float* __restrict__ be2, const float* __restrict__ m,
                       float* __restrict__ agg, int N) {
  __shared__ float aT[16][36];        // attention chunk, padded stride
  __shared__ float mT[2][32][HID];    // double-buffered m chunk (2 x 16 KB)
  __shared__ float xi[16][3];
  __shared__ float w1[16], bb1[16], w2[16];
  __shared__ float bb2;

  const int tid = threadIdx.x;
  const int i0 = blockIdx.x * 16;

  if (tid < 16) { w1[tid] = We1[tid]; bb1[tid] = be1[tid]; w2[tid] = We2[tid]; }
  if (tid == 16) bb2 = be2[0];
  if (tid < 48) xi[tid / 3][tid % 3] = pos[i0 * 3 + tid];

  const unsigned mT0 = (unsigned)(uintptr_t)&mT[0][0][0];
  const unsigned mT1 = (unsigned)(uintptr_t)&mT[1][0][0];

  // prologue: kick off chunk 0 into buffer 0 (4 x b128 per thread = 16 KB total)
#pragma unroll
  for (int r = 0; r < 4; ++r) {
    int f = tid + r * 256;                              // float4 index 0..1023
    async_copy_b128(mT0 + f * 16, (const char*)m + (size_t)f * 16);
  }
  __syncthreads();

  const int lane = tid & 31;
  const int half = lane >> 4;   // 0: K{0,1}, 1: K{2,3}
  const int lr   = lane & 15;   // row (A) / column (B,C,D)
  const int wv   = tid >> 5;    // wave id -> H-column tile

  v8f acc = {};

  int buf = 0;
  for (int jc = 0; jc < N; jc += 32, buf ^= 1) {
    // compute attention chunk (16 x 32): 2 entries per thread (pure VALU,
    // overlaps with the in-flight async copy of the *current* chunk's m data)
    for (int e = tid; e < 512; e += 256) {
      int ii = e & 15, jj = e >> 4;
      int j = jc + jj;
      float dx = xi[ii][0] - pos[3 * j + 0];
      float dy = xi[ii][1] - pos[3 * j + 1];
      float dz = xi[ii][2] - pos[3 * j + 2];
      float d2 = dx * dx + dy * dy + dz * dz;
      float ev = bb2;
#pragma unroll
      for (int k = 0; k < 16; ++k) ev += silu(d2 * w1[k] + bb1[k]) * w2[k];
      float a = sigm(ev);
      if (i0 + ii == j) a = 0.0f;          // zero self-edges
      aT[ii][jj] = a;
    }
    // issue async copy of the NEXT chunk into the other buffer, then wait for
    // the current chunk's copies (async loads retire in order: <=4 outstanding
    // means the previous 4 have landed in LDS).
    if (jc + 32 < N) {
      const char* src = (const char*)(m + (size_t)(jc + 32) * HID);
      unsigned dst = buf ? mT0 : mT1;
#pragma unroll
      for (int r = 0; r < 4; ++r) {
        int f = tid + r * 256;
        async_copy_b128(dst + f * 16, src + (size_t)f * 16);
      }
      wait_async_le4();
    } else {
      wait_async_le0();
    }
    __syncthreads();   // aT visible + every wave's m-chunk portion landed
#pragma unroll
    for (int ks = 0; ks < 32; ks += 4) {
      int k0 = ks + half * 2;
      v2f A; A.x = aT[lr][k0];              A.y = aT[lr][k0 + 1];
      v2f B; B.x = mT[buf][k0][wv * 16 + lr]; B.y = mT[buf][k0 + 1][wv * 16 + lr];
      acc = __builtin_amdgcn_wmma_f32_16x16x4_f32(false, A, false, B,
                                                  (short)0, acc, false, false);
    }
    __syncthreads();   // protect aT / mT[buf] before next iteration writes
  }
#pragma unroll
  for (int r = 0; r < 8; ++r) {
    int M = r + half * 8;
    agg[(size_t)(i0 + M) * HID + wv * 16 + lr] = acc[r];
  }
}

// ---------------------------------------------------------------------------
// K3: cf = silu((h+agg)@W_out + b_out); sq = ||cf||^2; fused type_net ->
// type_velocity written straight to d_out[3N..]. One block (128 thr)/atom.
// ---------------------------------------------------------------------------
__global__ void k_feat(const float* __restrict__ h, const float* __restrict__ agg,
                       const float* __restrict__ W_out, const float* __restrict__ b_out,
                       const float* __restrict__ Wt1, const float* __restrict__ bt1,
                       const float* __restrict__ Wt2, const float* __restrict__ bt2,
                       const float* __restrict__ ctxproj, float* __restrict__ cf,
                       float* __restrict__ sq, float* __restrict__ tvout) {
  __shared__ float sh[HID];
  __shared__ float shc[HID];
  __shared__ float red[HID];
  __shared__ float st[TYPE_HID];
  int i = blockIdx.x, d = threadIdx.x;
  float u = h[(size_t)i * HID + d] + agg[(size_t)i * HID + d];
  sh[d] = u;
  __syncthreads();
  float c = b_out[d];
  for (int k = 0; k < HID; ++k) c += sh[k] * W_out[k * HID + d];
  c = silu(c);
  cf[(size_t)i * HID + d] = c;
  shc[d] = c;
  red[d] = c * c;
  __syncthreads();
  for (int s = 64; s > 0; s >>= 1) {
    if (d < s) red[d] += red[d + s];
    __syncthreads();
  }
  if (d == 0) sq[i] = red[0];
  if (d < TYPE_HID) {
    float t1 = bt1[d] + ctxproj[d];
    for (int k = 0; k < HID; ++k) t1 += shc[k] * Wt1[k * TYPE_HID + d];
    st[d] = silu(t1);
  }
  __syncthreads();
  if (d < NUM_TYPES) {
    float tv = bt2[d];
    for (int k = 0; k < TYPE_HID; ++k) tv += st[k] * Wt2[k * NUM_TYPES + d];
    tvout[i * NUM_TYPES + d] = tv;
  }
}

// ---------------------------------------------------------------------------
// K5: gram tiles via fp32 WMMA (K=128 over H), then per-pair force weights in
// registers; wave-shuffle + LDS float-atomic reduction; coord_velocity out.
// 16 i-rows per block, 8 waves stride over the 128 j-tiles; next tile's cf
// rows are prefetched (global_prefetch_b8) while the current tile computes.
// ---------------------------------------------------------------------------
__global__ void k_force(const float* __restrict__ pos, const float* __restrict__ cf,
                        const float* __restrict__ sq, const float* __restrict__ good,
                        float* __restrict__ out, int N) {
  __shared__ float cfI[16][132];    // padded to avoid bank conflicts
  __shared__ float xi[16][3];
  __shared__ float sqI[16];
  __shared__ float gdI[16];
  __shared__ float rowAcc[16][4];   // {sumw, fx, fy, fz}

  const int tid = threadIdx.x;
  const int i0 = blockIdx.x * 16;

  for (int idx = tid; idx < 16 * HID; idx += 256)
    cfI[idx >> 7][idx & 127] = cf[(size_t)i0 * HID + idx];
  if (tid < 48) xi[tid / 3][tid % 3] = pos[i0 * 3 + tid];
  if (tid < 16) { sqI[tid] = sq[i0 + tid]; gdI[tid] = good[i0 + tid]; }
  if (tid < 64) rowAcc[tid >> 2][tid & 3] = 0.0f;
  __syncthreads();

  const int lane = tid & 31;
  const int half = lane >> 4;
  const int lr   = lane & 15;
  const int wv   = tid >> 5;

  float accW[8], accX[8], accY[8], accZ[8];
#pragma unroll
  for (int r = 0; r < 8; ++r) { accW[r] = 0.f; accX[r] = 0.f; accY[r] = 0.f; accZ[r] = 0.f; }

  // prefetch this wave's first tile
  if (wv * 16 + lr < N) __builtin_prefetch(cf + (size_t)(wv * 16 + lr) * HID, 0, 3);

  for (int jt = wv; jt < N / 16; jt += 8) {
    const int jc = jt * 16;
    const int j  = jc + lr;                 // this lane's j column
    const float* bj = cf + (size_t)j * HID;
    // prefetch next tile for this wave while we compute the current one
    if (jt + 8 < N / 16) __builtin_prefetch(cf + (size_t)((jt + 8) * 16 + lr) * HID, 0, 3);
    v8f g = {};
#pragma unroll 4
    for (int ks = 0; ks < HID; ks += 4) {
      int k0 = ks + half * 2;
      v2f A; A.x = cfI[lr][k0]; A.y = cfI[lr][k0 + 1];
      v2f B; B.x = bj[k0];      B.y = bj[k0 + 1];
      g = __builtin_amdgcn_wmma_f32_16x16x4_f32(false, A, false, B,
                                                (short)0, g, false, false);
    }
    float pjx = pos[3 * j + 0], pjy = pos[3 * j + 1], pjz = pos[3 * j + 2];
    float sqj = sq[j];
    float gdj = good[j];
#pragma unroll
    for (int r = 0; r < 8; ++r) {
      int M = r + half * 8;
      int gi = i0 + M;
      float dx = xi[M][0] - pjx, dy = xi[M][1] - pjy, dz = xi[M][2] - pjz;
      float d2 = dx * dx + dy * dy + dz * dz;
      float invd = 1.0f / (sqrtf(d2) + 1e-6f);
      float pn = sqrtf(fmaxf(sqI[M] + sqj + 2.0f * g[r], 0.0f));
      float w = (gdI[M] != 0.0f && gdj != 0.0f && gi != j) ? invd * pn : 0.0f;
      accW[r] += w; accX[r] += w * pjx; accY[r] += w * pjy; accZ[r] += w * pjz;
    }
  }
  // reduce each row accumulator across the 16 lanes of the half-wave
#pragma unroll
  for (int r = 0; r < 8; ++r) {
#pragma unroll
    for (int off = 1; off < 16; off <<= 1) {
      accW[r] += __shfl_xor(accW[r], off, 16);
      accX[r] += __shfl_xor(accX[r], off, 16);
      accY[r] += __shfl_xor(accY[r], off, 16);
      accZ[r] += __shfl_xor(accZ[r], off, 16);
    }
  }
  if (lr == 0) {
#pragma unroll
    for (int r = 0; r < 8; ++r) {
      int M = r + half * 8;
      atomicAdd(&rowAcc[M][0], accW[r]);
      atomicAdd(&rowAcc[M][1], accX[r]);
      atomicAdd(&rowAcc[M][2], accY[r]);
      atomicAdd(&rowAcc[M][3], accZ[r]);
    }
  }
  __syncthreads();
  if (tid < 16) {
    int gi = i0 + tid;
    float sw = rowAcc[tid][0];
    out[3 * gi + 0] = rowAcc[tid][1] - sw * xi[tid][0];
    out[3 * gi + 1] = rowAcc[tid][2] - sw * xi[tid][1];
    out[3 * gi + 2] = rowAcc[tid][3] - sw * xi[tid][2];
  }
}

// ---------------------------------------------------------------------------
extern "C" void kernel_launch(void* const* d_in, const int* in_sizes, int n_in,
                              void* d_out, int out_size, void* d_ws, size_t ws_size,
                              hipStream_t stream) {
  const float* pos     = (const float*)d_in[0];   // (N,3)
  const float* af      = (const float*)d_in[1];   // (N,5)
  const float* ctx     = (const float*)d_in[2];   // (1,256)
  const float* W_embed = (const float*)d_in[3];
  const float* b_embed = (const float*)d_in[4];
  const float* We1     = (const float*)d_in[5];
  const float* be1     = (const float*)d_in[6];
  const float* We2     = (const float*)d_in[7];
  const float* be2     = (const float*)d_in[8];
  const float* W_msg   = (const float*)d_in[9];
  const float* b_msg   = (const float*)d_in[10];
  const float* W_out   = (const float*)d_in[11];
  const float* b_out   = (const float*)d_in[12];
  const float* Wt1     = (const float*)d_in[13];
  const float* bt1     = (const float*)d_in[14];
  const float* Wt2     = (const float*)d_in[15];
  const float* bt2     = (const float*)d_in[16];

  const int N = in_sizes[0] / 3;                  // 2048

  float* ws   = (float*)d_ws;
  float* h    = ws;
  float* m    = h + (size_t)N * HID;
  float* agg  = m + (size_t)N * HID;
  float* cf   = agg + (size_t)N * HID;
  float* sq   = cf + (size_t)N * HID;
  float* good = sq + N;
  float* ctxp = good + N;

  float* out   = (float*)d_out;                   // coord_velocity (N,3)
  float* tvout = out + (size_t)3 * N;             // type_velocity  (N,5)

  k_ctx  <<<1, TYPE_HID, 0, stream>>>(ctx, Wt1, ctxp);
  k_embed<<<N, HID, 0, stream>>>(af, W_embed, b_embed, W_msg, b_msg, h, m, good);
  k_attn <<<N / 16, 256, 0, stream>>>(pos, We1, be1, We2, be2, m, agg, N);
  k_feat <<<N, HID, 0, stream>>>(h, agg, W_out, b_out, Wt1, bt1, Wt2, bt2, ctxp,
                                 cf, sq, tvout);
  k_force<<<N / 16, 256, 0, stream>>>(pos, cf, sq, good, out, N);
}